// GCN_84052509983599
// MI455X (gfx1250) — compile-verified
//
#include <hip/hip_runtime.h>
#include <math.h>

// ---------------------------------------------------------------------------
// GCN w/ edge sampling on MI455X (gfx1250, wave32).
// GEMMs: V_WMMA_F32_16X16X4_F32, one wave per 16-row x Fo output strip
//        (A loaded once per k-step, TN WMMAs issued per A load).
// Scatter: L2-resident fp32 global atomics, float4-vectorized.
// ---------------------------------------------------------------------------

typedef __attribute__((ext_vector_type(2))) float v2f;
typedef __attribute__((ext_vector_type(8))) float v8f;

__device__ __forceinline__ void atomicAddF32(float* p, float v) {
  // lowers to global_atomic_add_f32 (device scope, relaxed)
  __hip_atomic_fetch_add(p, v, __ATOMIC_RELAXED, __HIP_MEMORY_SCOPE_AGENT);
}

// ---------------------------- utility kernels ------------------------------

__global__ void k_zero4(float4* __restrict__ p, long n4) {
  long i = (long)blockIdx.x * blockDim.x + threadIdx.x;
  long st = (long)gridDim.x * blockDim.x;
  float4 z = {0.0f, 0.0f, 0.0f, 0.0f};
  for (; i < n4; i += st) p[i] = z;
}

// deg[dst[e]] += ew[e]  (ew == nullptr -> 1.0)
__global__ void k_deg(const int* __restrict__ dst, const float* __restrict__ ew,
                      float* __restrict__ deg, int E) {
  int e = blockIdx.x * blockDim.x + threadIdx.x;
  if (e >= E) return;
  float w = ew ? ew[e] : 1.0f;
  if (w != 0.0f) atomicAddF32(&deg[dst[e]], w);
}

// dinv = rsqrt(deg + 1)  (+1 = self loop)
__global__ void k_dinv(const float* __restrict__ deg, float* __restrict__ dinv,
                       int n) {
  int i = blockIdx.x * blockDim.x + threadIdx.x;
  if (i < n) dinv[i] = rsqrtf(deg[i] + 1.0f);
}

// norm[e] = dinv[src]*dinv[dst]*ew   (ew == nullptr -> 1.0)
__global__ void k_norm(const int* __restrict__ src, const int* __restrict__ dst,
                       const float* __restrict__ dinv,
                       const float* __restrict__ ew, float* __restrict__ norm,
                       int E) {
  int e = blockIdx.x * blockDim.x + threadIdx.x;
  if (e >= E) return;
  float w = ew ? ew[e] : 1.0f;
  norm[e] = dinv[src[e]] * dinv[dst[e]] * w;
}

// ------------------------------- WMMA GEMM ---------------------------------
// C[M,N] = A[M,K] @ B[K,N], fp32. One wave computes a 16 x (TN*16) strip:
// A (the lane-strided operand) is loaded once per k-step and reused across
// TN WMMAs; TN v8f accumulators (TN*8 VGPRs) stay resident.
// A 16x4 fp32 layout: lanes 0-15 row=l16 hold K=k..k+1, lanes 16-31 K=k+2..k+3.
// B 4x16 fp32 layout (symmetric). C/D: VGPR v -> row (v + 8*half), col l16.
template <int TN>
__global__ void k_gemm_strip(const float* __restrict__ A,
                             const float* __restrict__ B,
                             float* __restrict__ C, int M, int K, int N) {
  int lane = threadIdx.x & 31;
  int wave = threadIdx.x >> 5;
  int tm = blockIdx.x * (blockDim.x >> 5) + wave;
  if (tm >= (M >> 4)) return;  // wave-uniform exit: EXEC stays all-1s
  int hl = lane >> 4;          // lane half selects K offset +0 / +2
  int l16 = lane & 15;         // A: row within tile; B/C: column

  const float* arow = A + (size_t)(tm * 16 + l16) * K;
  v8f zero = {};
  v8f acc[TN];
#pragma unroll
  for (int t = 0; t < TN; ++t) acc[t] = zero;

  for (int k = 0; k < K; k += 4) {
    int kk = k + 2 * hl;
    v2f a;
    a.x = arow[kk];
    a.y = arow[kk + 1];
    const float* brow = B + (size_t)kk * N + l16;
#pragma unroll
    for (int t = 0; t < TN; ++t) {
      v2f b;
      b.x = brow[t * 16];      // row kk
      b.y = brow[t * 16 + N];  // row kk+1
      acc[t] = __builtin_amdgcn_wmma_f32_16x16x4_f32(
          false, a, false, b, (short)0, acc[t], false, false);
    }
  }

  float* crow = C + (size_t)(tm * 16 + 8 * hl) * N + l16;
#pragma unroll
  for (int t = 0; t < TN; ++t)
#pragma unroll
    for (int v = 0; v < 8; ++v) crow[(size_t)v * N + t * 16] = acc[t][v];
}

// ------------------------------- scatter -----------------------------------
// agg[dst[e]*F + f..f+3] += XW[src[e]*F + f..f+3] * norm[e]; F = 4<<lf4.
// One float4 (global_load_b128) + 4 atomics per thread: index loads and
// address math amortized 4x. XW + AGG (51 MB) are L2-resident (192 MB L2).
__global__ void k_scatter4(const float* __restrict__ XW,
                           const int* __restrict__ src,
                           const int* __restrict__ dst,
                           const float* __restrict__ norm,
                           float* __restrict__ agg, long total4, int lf4) {
  long i = (long)blockIdx.x * blockDim.x + threadIdx.x;
  long st = (long)gridDim.x * blockDim.x;
  int gm = (1 << lf4) - 1;
  for (; i < total4; i += st) {
    long e = i >> lf4;
    int g = (int)(i & gm);
    float nv = norm[e];
    if (nv != 0.0f) {
      int s = src[e];
      int d = dst[e];
      float4 v = ((const float4*)(XW + ((size_t)s << (lf4 + 2))))[g];
      float* out = agg + ((size_t)d << (lf4 + 2)) + 4 * g;
      atomicAddF32(out + 0, v.x * nv);
      atomicAddF32(out + 1, v.y * nv);
      atomicAddF32(out + 2, v.z * nv);
      atomicAddF32(out + 3, v.w * nv);
    }
  }
}

// out = agg + xw * dinv^2 (self loop) + bias, optional relu  (float4)
__global__ void k_finalize4(const float4* __restrict__ agg,
                            const float4* __restrict__ xw,
                            const float* __restrict__ dinv,
                            const float4* __restrict__ bias,
                            float4* __restrict__ out, long total4, int lf4,
                            int relu) {
  long i = (long)blockIdx.x * blockDim.x + threadIdx.x;
  if (i >= total4) return;
  long row = i >> lf4;
  int g = (int)(i & ((1 << lf4) - 1));
  float dv = dinv[row];
  float s = dv * dv;
  float4 a = agg[i], x = xw[i], b = bias[g];
  float4 v = {a.x + x.x * s + b.x, a.y + x.y * s + b.y, a.z + x.z * s + b.z,
              a.w + x.w * s + b.w};
  if (relu) {
    v.x = fmaxf(v.x, 0.0f);
    v.y = fmaxf(v.y, 0.0f);
    v.z = fmaxf(v.z, 0.0f);
    v.w = fmaxf(v.w, 0.0f);
  }
  out[i] = v;
}

// mask[e] = sigmoid(dot64(tz[src], tz[dst])) > rand[e]  (SAMPLING_RATIO = 1)
__global__ void k_mask(const float* __restrict__ tz, const int* __restrict__ src,
                       const int* __restrict__ dst, const float* __restrict__ er,
                       float* __restrict__ mask, int E) {
  int e = blockIdx.x * blockDim.x + threadIdx.x;
  if (e >= E) return;
  const float4* a = (const float4*)(tz + (size_t)src[e] * 64);
  const float4* b = (const float4*)(tz + (size_t)dst[e] * 64);
  float dot = 0.0f;
#pragma unroll
  for (int i = 0; i < 16; ++i) {
    float4 av = a[i], bv = b[i];
    dot += av.x * bv.x + av.y * bv.y + av.z * bv.z + av.w * bv.w;
  }
  float p = 1.0f / (1.0f + expf(-dot));
  mask[e] = (p > er[e]) ? 1.0f : 0.0f;
}

// ------------------------------- launcher ----------------------------------

extern "C" void kernel_launch(void* const* d_in, const int* in_sizes, int n_in,
                              void* d_out, int out_size, void* d_ws,
                              size_t ws_size, hipStream_t stream) {
  const float* x  = (const float*)d_in[0];
  const int*   ei = (const int*)d_in[1];
  const float* er = (const float*)d_in[2];
  const float* w0 = (const float*)d_in[3];
  const float* b0 = (const float*)d_in[4];
  const float* w1 = (const float*)d_in[5];
  const float* b1 = (const float*)d_in[6];
  const float* w2 = (const float*)d_in[7];
  const float* b2 = (const float*)d_in[8];

  const int FIN = 128, HID = 128, FOUT = 64;
  const int N = in_sizes[0] / FIN;  // 50000 (multiple of 16)
  const int E = in_sizes[1] / 2;    // 800000
  const int* src = ei;
  const int* dstv = ei + E;

  float* ws = (float*)d_ws;
  float* XW    = ws;                          // N*128
  float* AGG   = XW + (size_t)N * HID;        // N*128
  float* H0    = AGG + (size_t)N * HID;       // N*128
  float* H1    = H0 + (size_t)N * HID;        // N*128
  float* TMPZ  = H1 + (size_t)N * HID;        // N*64
  float* DEG   = TMPZ + (size_t)N * FOUT;     // N (reused for both passes)
  float* DINV1 = DEG + N;                     // N
  float* DINVM = DINV1 + N;                   // N
  float* NORM1 = DINVM + N;                   // E
  float* NORMM = NORM1 + E;                   // E
  float* MASK  = NORMM + E;                   // E

  const int T = 256;
  auto nb = [](long n) { return (int)((n + 255) / 256); };

  auto conv = [&](const float* xin, const float* W, const float* bias, int Kin,
                  int Fo, const float* norm, const float* dinv, float* out,
                  int relu, int lf) {
    int strips = N / 16;  // one wave per 16-row strip, 8 waves per block
    if (Fo == 128)
      k_gemm_strip<8><<<(strips + 7) / 8, 256, 0, stream>>>(xin, W, XW, N, Kin,
                                                            Fo);
    else
      k_gemm_strip<4><<<(strips + 7) / 8, 256, 0, stream>>>(xin, W, XW, N, Kin,
                                                            Fo);
    long n4 = (long)N * Fo / 4;
    long t4 = (long)E * Fo / 4;
    int lf4 = lf - 2;
    k_zero4<<<nb(n4), T, 0, stream>>>((float4*)AGG, n4);
    k_scatter4<<<nb(t4), T, 0, stream>>>(XW, src, dstv, norm, AGG, t4, lf4);
    k_finalize4<<<nb(n4), T, 0, stream>>>((const float4*)AGG,
                                          (const float4*)XW, dinv,
                                          (const float4*)bias, (float4*)out,
                                          n4, lf4, relu);
  };

  // ---- degrees / norms for the all-ones edge weights ----
  k_zero4<<<nb(N / 4), T, 0, stream>>>((float4*)DEG, N / 4);
  k_deg<<<nb(E), T, 0, stream>>>(dstv, nullptr, DEG, E);
  k_dinv<<<nb(N), T, 0, stream>>>(DEG, DINV1, N);
  k_norm<<<nb(E), T, 0, stream>>>(src, dstv, DINV1, nullptr, NORM1, E);

  // ---- pass 1 (full graph) -> TMPZ ----
  conv(x,  w0, b0, FIN, HID,  NORM1, DINV1, H0,   1, 7);
  conv(H0, w1, b1, HID, HID,  NORM1, DINV1, H1,   1, 7);
  conv(H1, w2, b2, HID, FOUT, NORM1, DINV1, TMPZ, 0, 6);

  // ---- edge sampling mask ----
  k_mask<<<nb(E), T, 0, stream>>>(TMPZ, src, dstv, er, MASK, E);

  // ---- degrees / norms for masked edges ----
  k_zero4<<<nb(N / 4), T, 0, stream>>>((float4*)DEG, N / 4);
  k_deg<<<nb(E), T, 0, stream>>>(dstv, MASK, DEG, E);
  k_dinv<<<nb(N), T, 0, stream>>>(DEG, DINVM, N);
  k_norm<<<nb(E), T, 0, stream>>>(src, dstv, DINVM, MASK, NORMM, E);

  // ---- pass 2 (sampled hidden layers, full last layer) -> d_out ----
  conv(x,  w0, b0, FIN, HID,  NORMM, DINVM, H0, 1, 7);
  conv(H0, w1, b1, HID, HID,  NORMM, DINVM, H1, 1, 7);
  conv(H1, w2, b2, HID, FOUT, NORM1, DINV1, (float*)d_out, 0, 6);
}